// CurvatureWeightedBoundaryLoss_84937273246176
// MI455X (gfx1250) — compile-verified
//
#include <hip/hip_runtime.h>
#include <hip/hip_bf16.h>

#define INFV 1.0e10f
#define HH 256
#define WW 256
#define NB 8
#define NCLS 4
#define NPAIR (NB * (NCLS - 1)) /* 24 (b, c) pairs */
#define NMAP (NPAIR * 2)        /* 48 distance maps  */
#define HW (HH * WW)            /* 65536 */

typedef __attribute__((ext_vector_type(2))) float v2f;
typedef __attribute__((ext_vector_type(8))) float v8f;

// ---------------------------------------------------------------------------
// Pass 1: row transform of f = (site ? 0 : INF) for both polarities at once.
//   gT[pair*2+s][j][i] = min_k f_s[i][k] + (j-k)^2      (output transposed)
// Block: (tile of 16 image rows) x (one (b,c) pair). Thread = output coord j.
// ---------------------------------------------------------------------------
__global__ void edt_pass1(const int* __restrict__ target, float* __restrict__ gT) {
  const int pair = blockIdx.y;          // 0..23
  const int b = pair / 3;
  const int c = pair % 3 + 1;
  const int x0 = blockIdx.x * 16;       // input row group
  const int tid = threadIdx.x;          // 0..255

  __shared__ float mask[16 * WW];
  const int* tb = target + b * HW;
  for (int e = tid; e < 16 * WW; e += 256) {
    int row = e >> 8, col = e & 255;
    mask[e] = (tb[(x0 + row) * WW + col] == c) ? 1.0f : 0.0f;
  }
  __syncthreads();

  const int y = tid;                    // transformed coordinate j
  const float fy = (float)y;
  float* __restrict__ g0 = gT + (size_t)(pair * 2 + 0) * HW;
  float* __restrict__ g1 = gT + (size_t)(pair * 2 + 1) * HW;

  for (int xq = 0; xq < 4; ++xq) {
    float4 o0, o1;
    #pragma unroll
    for (int xi = 0; xi < 4; ++xi) {
      const int xx = xq * 4 + xi;
      const float* __restrict__ mrow = &mask[xx * WW];
      float d0 = INFV, d1 = INFV;
      #pragma unroll 4
      for (int k = 0; k < WW; ++k) {
        float dj = fy - (float)k;
        float dj2 = dj * dj;
        float m = mrow[k];              // LDS broadcast (uniform address)
        d0 = fminf(d0, (m != 0.0f) ? dj2 : INFV);
        d1 = fminf(d1, (m != 0.0f) ? INFV : dj2);
      }
      ((float*)&o0)[xi] = d0;
      ((float*)&o1)[xi] = d1;
    }
    const int x = x0 + xq * 4;
    *(float4*)&g0[y * WW + x] = o0;     // transposed write: [j][i]
    *(float4*)&g1[y * WW + x] = o1;
  }
}

// ---------------------------------------------------------------------------
// Pass 2: generic min-plus row transform, output transposed again:
//   D[m][y][x] = min_k A[m][x][k] + (y-k)^2   -> D ends up as d2[i][j].
// ---------------------------------------------------------------------------
__global__ void edt_pass2(const float* __restrict__ A, float* __restrict__ D) {
  const int m = blockIdx.y;             // 0..47
  const int x0 = blockIdx.x * 16;
  const int tid = threadIdx.x;

  __shared__ float a[16 * WW];          // 16 input rows, 16 KB
  const float* __restrict__ Am = A + (size_t)m * HW;
  for (int e = tid; e < 16 * WW; e += 256) {
    int row = e >> 8, col = e & 255;
    a[e] = Am[(x0 + row) * WW + col];   // coalesced row reads
  }
  __syncthreads();

  const int y = tid;
  const float fy = (float)y;
  float dd[16];
  #pragma unroll
  for (int i = 0; i < 16; ++i) dd[i] = INFV;

  for (int k = 0; k < WW; ++k) {
    float dj = fy - (float)k;
    float dj2 = dj * dj;                // shared across the 16-wide x-tile
    #pragma unroll
    for (int xi = 0; xi < 16; ++xi)
      dd[xi] = fminf(dd[xi], a[xi * WW + k] + dj2);
  }

  float* __restrict__ Dm = D + (size_t)m * HW;
  #pragma unroll
  for (int q = 0; q < 4; ++q) {
    float4 o = {dd[q * 4 + 0], dd[q * 4 + 1], dd[q * 4 + 2], dd[q * 4 + 3]};
    *(float4*)&Dm[y * WW + x0 + q * 4] = o;
  }
}

// ---------------------------------------------------------------------------
// Per-(b,c) site counts for the degenerate-mask test (integer => deterministic)
// ---------------------------------------------------------------------------
__global__ void count_kernel(const int* __restrict__ target, int* __restrict__ counts) {
  const int pair = blockIdx.x;
  const int b = pair / 3;
  const int c = pair % 3 + 1;
  const int* tb = target + b * HW;
  int cnt = 0;
  for (int e = threadIdx.x; e < HW; e += 256) cnt += (tb[e] == c) ? 1 : 0;
  __shared__ int sh[256];
  sh[threadIdx.x] = cnt;
  __syncthreads();
  for (int s = 128; s > 0; s >>= 1) {
    if (threadIdx.x < s) sh[threadIdx.x] += sh[threadIdx.x + s];
    __syncthreads();
  }
  if (threadIdx.x == 0) counts[pair] = sh[0];
}

// ---------------------------------------------------------------------------
// Fused softmax + |p-t| * w * (sqrt(d2m)+sqrt(d2n)) with WMMA-based wave sum.
// 1024 blocks x 256 threads, exactly 2 pixels/thread -> EXEC stays all-ones.
// ---------------------------------------------------------------------------
__global__ void combine_kernel(const float* __restrict__ pred,
                               const int* __restrict__ target,
                               const float* __restrict__ w,
                               const float* __restrict__ d2,
                               const int* __restrict__ counts,
                               float* __restrict__ partial) {
  const int tid = threadIdx.x;
  const int gid = blockIdx.x * 256 + tid;

  float acc = 0.0f;
  #pragma unroll
  for (int it = 0; it < 2; ++it) {
    const int p = gid + it * (1024 * 256);  // pixel id in [0, 524288)
    const int b = p >> 16;
    const int ij = p & 65535;

    const float* pb = pred + (size_t)b * NCLS * HW + ij;
    float l0 = pb[0], l1 = pb[HW], l2 = pb[2 * HW], l3 = pb[3 * HW];
    float mx = fmaxf(fmaxf(l0, l1), fmaxf(l2, l3));
    float e0 = expf(l0 - mx), e1 = expf(l1 - mx);
    float e2 = expf(l2 - mx), e3 = expf(l3 - mx);
    float inv = 1.0f / (e0 + e1 + e2 + e3);
    float probs[3] = {e1 * inv, e2 * inv, e3 * inv};

    const int t = target[p];
    const float ww = w[p];
    #pragma unroll
    for (int ci = 0; ci < 3; ++ci) {
      const int c = ci + 1;
      const int pair = b * 3 + ci;
      float tc = (t == c) ? 1.0f : 0.0f;
      float err = fabsf(probs[ci] - tc);
      float s0 = sqrtf(d2[(size_t)(pair * 2 + 0) * HW + ij]);
      float s1 = sqrtf(d2[(size_t)(pair * 2 + 1) * HW + ij]);
      int cnt = counts[pair];
      float dist = (cnt != 0 && cnt != HW) ? (s0 + s1) : 0.0f; // branchless select
      acc += err * ww * dist;
    }
  }

  // Wave-level partial sum on the matrix pipe: A = ones(16x4), B vgpr0 = acc,
  // vgpr1 = 0  =>  d[0] lane n (n<16) = acc[n] + acc[n+16]  (layout-robust).
  v2f aones = {1.0f, 1.0f};
  v2f bv = {acc, 0.0f};
  v8f cz = {};
  v8f dsum = __builtin_amdgcn_wmma_f32_16x16x4_f32(
      /*neg_a=*/false, aones, /*neg_b=*/false, bv,
      /*c_mod=*/(short)0, cz, /*reuse_a=*/false, /*reuse_b=*/false);

  __shared__ float wred[8 * 16];
  const int wave = tid >> 5, lane = tid & 31;
  if (lane < 16) wred[wave * 16 + lane] = dsum[0];
  __syncthreads();
  if (tid == 0) {
    float s = 0.0f;
    for (int i = 0; i < 8 * 16; ++i) s += wred[i];  // fixed order: deterministic
    partial[blockIdx.x] = s;
  }
}

// ---------------------------------------------------------------------------
// Deterministic final reduction of block partials -> scalar loss.
// ---------------------------------------------------------------------------
__global__ void final_kernel(const float* __restrict__ partial, int n,
                             float* __restrict__ out, float scale) {
  __shared__ float sh[256];
  float s = 0.0f;
  for (int i = threadIdx.x; i < n; i += 256) s += partial[i];
  sh[threadIdx.x] = s;
  __syncthreads();
  if (threadIdx.x == 0) {
    float t = 0.0f;
    for (int i = 0; i < 256; ++i) t += sh[i];
    out[0] = t * scale;
  }
}

extern "C" void kernel_launch(void* const* d_in, const int* in_sizes, int n_in,
                              void* d_out, int out_size, void* d_ws, size_t ws_size,
                              hipStream_t stream) {
  const float* pred   = (const float*)d_in[0];   // [8,4,256,256] f32
  const int*   target = (const int*)d_in[1];     // [8,256,256]   i32
  const float* bw     = (const float*)d_in[2];   // [8,1,256,256] f32
  float* out = (float*)d_out;

  // Workspace: gT (48 maps) | d2 (48 maps) | counts (24 ints, padded) | partials
  float* gT = (float*)d_ws;
  float* d2 = gT + (size_t)NMAP * HW;
  int* counts = (int*)(d2 + (size_t)NMAP * HW);
  float* partial = (float*)(counts + 32);

  dim3 blk(256, 1, 1);
  edt_pass1<<<dim3(16, NPAIR), blk, 0, stream>>>(target, gT);
  count_kernel<<<dim3(NPAIR), blk, 0, stream>>>(target, counts);
  edt_pass2<<<dim3(16, NMAP), blk, 0, stream>>>(gT, d2);
  combine_kernel<<<dim3(1024), blk, 0, stream>>>(pred, target, bw, d2, counts, partial);
  final_kernel<<<dim3(1), blk, 0, stream>>>(partial, 1024, out,
                                            1.0f / (3.0f * (float)(NB * HW)));
}